// CTC_Loss_Module_52115133169664
// MI455X (gfx1250) — compile-verified
//
#include <hip/hip_runtime.h>
#include <stdint.h>

// CTC loss forward, CDNA5 (gfx1250).
// T=512, B=64, V=1296, S=30, L=61. One wave32 per batch element; alpha held in
// registers (2 lattice cells per lane, neighbor exchange via lane permutes).
// Emission gathers (31 values per timestep) are pipelined PF=6 deep through the
// CDNA5 async engine: global_load_async_to_lds_b32 -> s_wait_asynccnt.

#define T_DIM 512
#define B_DIM 64
#define V_DIM 1296
#define S_DIM 30
#define L_DIM 61          // 2*S + 1
#define NEGF  (-1e30f)
#define PF    6           // prefetch depth (outstanding async gathers)
#define NBUF  8           // LDS ring-buffer slots (power of 2, >= PF)

__device__ __forceinline__ float lse2(float x, float y) {
    // logaddexp; safe with NEGF surrogate (-1e30): exp underflows to 0.
    float mx = fmaxf(x, y);
    float mn = fminf(x, y);
    return mx + log1pf(__expf(mn - mx));
}

__global__ __launch_bounds__(32) void ctc_alpha_kernel(
    const float* __restrict__ log_probs,     // (T, B, V)
    const int*   __restrict__ targets,       // (B*S)
    const int*   __restrict__ input_lengths, // (B)
    float*       __restrict__ loss_ws)       // (B) per-sample loss out
{
    __shared__ float buf[NBUF][32];          // async gather landing ring

    const int b    = blockIdx.x;
    const int lane = threadIdx.x;            // wave32
    const int len  = input_lengths[b];       // uniform per block

    // Gathered value j per timestep: j==0 -> blank(0), j=1..30 -> targets[j-1].
    const int gidx = (lane >= 1 && lane <= 30) ? targets[b * S_DIM + lane - 1] : 0;

    // Global byte address of this lane's gather at t=0; advances by B*V*4 per t.
    const uint64_t tstep  = 4ull * B_DIM * V_DIM;
    const uint64_t gaddr0 = (uint64_t)(uintptr_t)log_probs
                          + 4ull * ((uint64_t)b * V_DIM + (uint64_t)gidx);
    // Wave-relative LDS byte offset of this lane's slot-0 cell (low 32 bits of
    // the generic LDS pointer are the workgroup-relative offset).
    const uint32_t lds0 = (uint32_t)(uintptr_t)&buf[0][lane];

    // Prime the async pipeline: PF outstanding gathers (ASYNCcnt = PF).
    #pragma unroll
    for (int tt = 0; tt < PF; ++tt) {
        uint32_t ldsa = lds0 + (uint32_t)((tt & (NBUF - 1)) * 32 * 4);
        uint64_t ga   = gaddr0 + (uint64_t)tt * tstep;
        asm volatile("global_load_async_to_lds_b32 %0, %1, off"
                     :: "v"(ldsa), "v"(ga) : "memory");
    }

    // Per-lane lattice metadata: positions l_lo = lane, l_hi = lane + 32.
    const int  l_lo     = lane;
    const int  l_hi     = lane + 32;
    const bool hi_valid = (l_hi < L_DIM);    // lanes 0..28 carry a hi cell

    int bidx_lo = (l_lo & 1) ? ((l_lo + 1) >> 1) : 0;
    int bidx_hi = (l_hi & 1) ? ((l_hi + 1) >> 1) : 0;
    if (bidx_hi > 31) bidx_hi = 31;          // clamp dead lanes into the buffer

    bool allow_lo = false, allow_hi = false; // skip-transition legality
    if (l_lo & 1) {
        int k  = (l_lo - 1) >> 1;
        int e  = targets[b * S_DIM + k];
        int e2 = (l_lo >= 3) ? targets[b * S_DIM + k - 1] : 0;
        allow_lo = (e != e2);                // e >= 1 always (blank excluded)
    }
    if ((l_hi & 1) && hi_valid) {
        int k  = (l_hi - 1) >> 1;
        int e  = targets[b * S_DIM + k];
        int e2 = targets[b * S_DIM + k - 1];
        allow_hi = (e != e2);
    }

    // ---- t = 0 (peeled): alpha0 from lp(0) ----
    asm volatile("s_wait_asynccnt %0" :: "i"(PF - 1) : "memory");
    volatile float* bt0 = &buf[0][0];
    float a_lo = (lane <= 1) ? bt0[bidx_lo] : NEGF; // l=0 blank, l=1 first label
    float a_hi = NEGF;
    {
        uint32_t ldsa = lds0 + (uint32_t)((PF & (NBUF - 1)) * 32 * 4);
        uint64_t ga   = gaddr0 + (uint64_t)PF * tstep;
        asm volatile("global_load_async_to_lds_b32 %0, %1, off"
                     :: "v"(ldsa), "v"(ga) : "memory");
    }

    for (int t = 1; t < T_DIM; ++t) {
        // Gather for step t is complete once at most PF-1 async ops remain
        // outstanding (async loads retire in order).
        asm volatile("s_wait_asynccnt %0" :: "i"(PF - 1) : "memory");
        volatile float* bt = &buf[t & (NBUF - 1)][0];
        float lp_lo = bt[bidx_lo];
        float lp_hi = bt[bidx_hi];

        float p1_lo = __shfl_up(a_lo, 1, 32);
        float p2_lo = __shfl_up(a_lo, 2, 32);
        float p1_hi = __shfl_up(a_hi, 1, 32);
        float p2_hi = __shfl_up(a_hi, 2, 32);
        float top_m1 = __shfl(a_lo, 31, 32); // alpha[31]
        float top_m2 = __shfl(a_lo, 30, 32); // alpha[30]
        if (lane == 0) { p1_lo = NEGF; p2_lo = NEGF; p1_hi = top_m1; p2_hi = top_m2; }
        if (lane == 1) { p2_lo = NEGF; p2_hi = top_m1; }

        float m_lo = lse2(a_lo, p1_lo);
        if (allow_lo) m_lo = lse2(m_lo, p2_lo);
        float m_hi = lse2(a_hi, p1_hi);
        if (allow_hi) m_hi = lse2(m_hi, p2_hi);
        a_lo = m_lo + lp_lo;
        a_hi = hi_valid ? (m_hi + lp_hi) : NEGF;

        if (t == len - 1) {                      // uniform branch per wave
            float v59 = __shfl(a_hi, 27, 32);    // l = 27+32 = 59 (last label)
            float v60 = __shfl(a_hi, 28, 32);    // l = 28+32 = 60 (last blank)
            float lb = -lse2(v59, v60);
            if (lb >= 1e29f) lb = 0.0f;          // zero_infinity
            if (lane == 0) loss_ws[b] = lb;
            break;
        }

        // Refill the pipeline: issue gather for t+PF into its ring slot. The
        // value from this iteration's slot is already consumed above, and the
        // slot written here ((t+PF)&7) was last read NBUF-PF (2) iterations ago.
        int tn = t + PF;
        if (tn < T_DIM) {
            uint32_t ldsa = lds0 + (uint32_t)((tn & (NBUF - 1)) * 32 * 4);
            uint64_t ga   = gaddr0 + (uint64_t)tn * tstep;
            asm volatile("global_load_async_to_lds_b32 %0, %1, off"
                         :: "v"(ldsa), "v"(ga) : "memory");
        }
    }
    // S_ENDPGM implicitly waits for any still-outstanding async ops.
}

// Deterministic fixed-order reduction: mean_b( loss_b / target_len_b ).
__global__ void ctc_reduce_kernel(const float* __restrict__ loss_ws,
                                  const int*   __restrict__ target_lengths,
                                  float*       __restrict__ out)
{
    if (blockIdx.x == 0 && threadIdx.x == 0) {
        float s = 0.0f;
        for (int b = 0; b < B_DIM; ++b)
            s += loss_ws[b] / (float)target_lengths[b];
        out[0] = s / (float)B_DIM;
    }
}

extern "C" void kernel_launch(void* const* d_in, const int* in_sizes, int n_in,
                              void* d_out, int out_size, void* d_ws, size_t ws_size,
                              hipStream_t stream) {
    const float* log_probs      = (const float*)d_in[0]; // (T*B*V) f32
    const int*   targets        = (const int*)  d_in[1]; // (B*S)
    const int*   input_lengths  = (const int*)  d_in[2]; // (B)
    const int*   target_lengths = (const int*)  d_in[3]; // (B)
    float*       out            = (float*)d_out;         // scalar
    float*       ws             = (float*)d_ws;          // >= B floats

    ctc_alpha_kernel<<<B_DIM, 32, 0, stream>>>(log_probs, targets, input_lengths, ws);
    ctc_reduce_kernel<<<1, 32, 0, stream>>>(ws, target_lengths, out);
}